// Attention_2765958939361
// MI455X (gfx1250) — compile-verified
//
#include <hip/hip_runtime.h>
#include <hip/hip_bf16.h>
#include <math.h>

// ---------------------------------------------------------------------------
// Decode attention, fp32 end-to-end (memory-bound: ~2.4 GB / ~3.2 GFLOP).
//   BSZ=16, SEQLEN=1, EMB=4096, N_HEADS=32, HEAD_DIM=128, kv_len=4096
// ---------------------------------------------------------------------------
#define EMB      4096
#define BSZ      16
#define NHEADS   32
#define HDIM     128
#define KVLEN    4096
#define NCHUNK   8
#define CHUNK    512
#define KHALF    2048                    // K split for GEMM parallelism
#define SCALE    0.08838834764831845f    // 1/sqrt(128)

typedef __attribute__((ext_vector_type(2))) float v2f;
typedef __attribute__((ext_vector_type(4))) float v4f;
typedef __attribute__((ext_vector_type(8))) float v8f;

#define V2LO(v) __builtin_shufflevector((v), (v), 0, 1)
#define V2HI(v) __builtin_shufflevector((v), (v), 2, 3)

__device__ __forceinline__ float wave_sum(float v) {
    v += __shfl_xor(v, 16, 32);
    v += __shfl_xor(v, 8, 32);
    v += __shfl_xor(v, 4, 32);
    v += __shfl_xor(v, 2, 32);
    v += __shfl_xor(v, 1, 32);
    return v;
}
__device__ __forceinline__ float wave_max(float v) {
    v = fmaxf(v, __shfl_xor(v, 16, 32));
    v = fmaxf(v, __shfl_xor(v, 8, 32));
    v = fmaxf(v, __shfl_xor(v, 4, 32));
    v = fmaxf(v, __shfl_xor(v, 2, 32));
    v = fmaxf(v, __shfl_xor(v, 1, 32));
    return v;
}

// ---------------------------------------------------------------------------
// Shared GEMM body: one wave computes a 16(M) x 16(N) strip over one K-half
// with V_WMMA_F32_16X16X4_F32, b128 loads, NT policy on the weight stream.
// K-column remap trick: lane half h loads contiguous cols (k + 4h .. k+4h+3);
// WMMA #0 consumes elems (0,1), WMMA #1 elems (2,3). A and B use the same
// permutation of global k -> hardware K-slot, so the accumulated sum is exact.
// ---------------------------------------------------------------------------
__device__ __forceinline__ v8f gemm_strip(const float* __restrict__ a_row,
                                          const float* __restrict__ b_row) {
    v8f acc0 = {}, acc1 = {}, acc2 = {}, acc3 = {};
    const float* pa = a_row;
    const float* pb = b_row;
    for (int k = 0; k < KHALF; k += 16) {
        v4f a0 = *(const v4f*)(pa);
        v4f b0 = __builtin_nontemporal_load((const v4f*)(pb));
        v4f a1 = *(const v4f*)(pa + 8);
        v4f b1 = __builtin_nontemporal_load((const v4f*)(pb + 8));
        acc0 = __builtin_amdgcn_wmma_f32_16x16x4_f32(false, V2LO(a0), false, V2LO(b0),
                                                     (short)0, acc0, false, false);
        acc1 = __builtin_amdgcn_wmma_f32_16x16x4_f32(false, V2HI(a0), false, V2HI(b0),
                                                     (short)0, acc1, false, false);
        acc2 = __builtin_amdgcn_wmma_f32_16x16x4_f32(false, V2LO(a1), false, V2LO(b1),
                                                     (short)0, acc2, false, false);
        acc3 = __builtin_amdgcn_wmma_f32_16x16x4_f32(false, V2HI(a1), false, V2HI(b1),
                                                     (short)0, acc3, false, false);
        pa += 16;
        pb += 16;
    }
    return (acc0 + acc1) + (acc2 + acc3);
}

// ---------------------------------------------------------------------------
// Kernel 1: QKV projection partials. Grid: 3 wsel x 2 khalf x 32 nblk = 192
// blocks, 256 threads (8 waves). Raw fp32 partials -> ws_qkvp.
// ---------------------------------------------------------------------------
__global__ void qkv_gemm(const float* __restrict__ x,
                         const float* __restrict__ wq,
                         const float* __restrict__ wk,
                         const float* __restrict__ wv,
                         float* __restrict__ qkvp) {
    const int wg    = blockIdx.x;
    const int wsel  = wg >> 6;           // /64 : 0=q 1=k 2=v
    const int rem   = wg & 63;
    const int khalf = rem >> 5;
    const int nblk  = rem & 31;
    const int wave  = threadIdx.x >> 5;
    const int lane  = threadIdx.x & 31;
    const int n0    = nblk * 128 + wave * 16;
    const int half  = lane >> 4;
    const int mr    = lane & 15;
    const int k0    = khalf * KHALF;

    const float* w = (wsel == 0) ? wq : ((wsel == 1) ? wk : wv);

    v8f acc = gemm_strip(x + (size_t)mr * EMB + k0 + 4 * half,
                         w + (size_t)(n0 + mr) * EMB + k0 + 4 * half);

    // C layout: lane l, vgpr j -> M = j + 8*(l>>4), N = n0 + (l&15)
    float* dst = qkvp + (size_t)(wsel * 2 + khalf) * (BSZ * EMB);
    const int n = n0 + mr;
#pragma unroll
    for (int j = 0; j < 8; ++j) {
        const int M = j + 8 * half;
        dst[(size_t)M * EMB + n] = acc[j];
    }
}

// ---------------------------------------------------------------------------
// Kernel 2: combine K-halves + RoPE (q,k) -> ws_q / ws_k / ws_v.
// Grid: 98304 pairs / 256 = 384 blocks. Each thread owns one (even,odd) pair.
// ---------------------------------------------------------------------------
__global__ void qkv_rope_reduce(const float* __restrict__ qkvp,
                                const float* __restrict__ fcos,
                                const float* __restrict__ fsin,
                                float* __restrict__ ws_q,
                                float* __restrict__ ws_k,
                                float* __restrict__ ws_v) {
    const int e    = blockIdx.x * 256 + threadIdx.x;  // 0 .. 98303
    const int wsel = e >> 15;                          // /32768
    const int pe   = e & 32767;
    const int m    = pe >> 11;                         // /2048
    const int n    = (pe & 2047) * 2;

    const float* p0 = qkvp + (size_t)(wsel * 2 + 0) * (BSZ * EMB) + (size_t)m * EMB;
    const float* p1 = qkvp + (size_t)(wsel * 2 + 1) * (BSZ * EMB) + (size_t)m * EMB;
    float r = p0[n] + p1[n];
    float i = p0[n + 1] + p1[n + 1];

    if (wsel != 2) {
        const int   fi = (n & (HDIM - 1)) >> 1;
        const float c  = fcos[fi];
        const float s  = fsin[fi];
        const float rr = r * c - i * s;
        const float ii = r * s + i * c;
        r = rr;
        i = ii;
    }
    float* dst = (wsel == 0) ? ws_q : ((wsel == 1) ? ws_k : ws_v);
    dst[(size_t)m * EMB + n]     = r;
    dst[(size_t)m * EMB + n + 1] = i;
}

// ---------------------------------------------------------------------------
// Kernel 3: flash-decode partials. One block = (b, h, kv-chunk of 512).
// Grid: 16*32*8 = 4096 blocks, 256 threads (8 waves). NT loads on the
// streamed 2.15 GB of KV cache. Fresh row (t == start_pos) from ws_k/ws_v.
// ---------------------------------------------------------------------------
__global__ void attn_partial(const float* __restrict__ cache_k,
                             const float* __restrict__ cache_v,
                             const float* __restrict__ ws_q,
                             const float* __restrict__ ws_k,
                             const float* __restrict__ ws_v,
                             float* __restrict__ part,
                             const int* __restrict__ sp) {
    const int blk  = blockIdx.x;
    const int b    = blk >> 8;
    const int rem  = blk & 255;
    const int h    = rem >> 3;
    const int c    = rem & 7;
    const int tid  = threadIdx.x;
    const int wave = tid >> 5;
    const int lane = tid & 31;
    const int spos = sp[0];

    __shared__ __align__(16) float q_s[HDIM];
    __shared__ float s_s[CHUNK];
    __shared__ float acc_s[8 * HDIM];
    __shared__ float red_s[8];
    __shared__ float mshare, lshare;

    if (tid < HDIM) q_s[tid] = ws_q[(size_t)b * EMB + h * HDIM + tid];
    __syncthreads();

    const v4f qv = *reinterpret_cast<const v4f*>(&q_s[lane * 4]);

    // cache layout [b][t][h][d]
    const float* kb = cache_k + (size_t)b * KVLEN * (NHEADS * HDIM) + (size_t)h * HDIM;
    const float* vb = cache_v + (size_t)b * KVLEN * (NHEADS * HDIM) + (size_t)h * HDIM;
    const float* fk = ws_k + (size_t)b * EMB + h * HDIM;
    const float* fv = ws_v + (size_t)b * EMB + h * HDIM;

    const int t0 = c * CHUNK + wave * 64;

    // ---- scores: one coalesced 512B row (b128/lane, NT) per t ----
    for (int i = 0; i < 64; ++i) {
        const int    t  = t0 + i;
        const float* kr = (t == spos) ? fk : (kb + (size_t)t * (NHEADS * HDIM));
        const v4f    kv = __builtin_nontemporal_load((const v4f*)(kr + lane * 4));
        float d = qv[0] * kv[0] + qv[1] * kv[1] + qv[2] * kv[2] + qv[3] * kv[3];
        d = wave_sum(d);
        if (lane == 0) s_s[wave * 64 + i] = d * SCALE;
    }
    __syncthreads();

    // ---- chunk max ----
    float vm = fmaxf(s_s[tid], s_s[tid + 256]);
    vm = wave_max(vm);
    if (lane == 0) red_s[wave] = vm;
    __syncthreads();
    if (tid == 0) {
        float m = red_s[0];
        for (int i = 1; i < 8; ++i) m = fmaxf(m, red_s[i]);
        mshare = m;
    }
    __syncthreads();
    const float m = mshare;

    // ---- probs + chunk sum ----
    const float p0 = __expf(s_s[tid] - m);
    const float p1 = __expf(s_s[tid + 256] - m);
    s_s[tid]       = p0;
    s_s[tid + 256] = p1;
    float ls = wave_sum(p0 + p1);
    if (lane == 0) red_s[wave] = ls;
    __syncthreads();
    if (tid == 0) {
        float l = 0.0f;
        for (int i = 0; i < 8; ++i) l += red_s[i];
        lshare = l;
    }
    __syncthreads();

    // ---- probs @ V ----
    v4f acc = {};
    for (int i = 0; i < 64; ++i) {
        const int    t  = t0 + i;
        const float* vr = (t == spos) ? fv : (vb + (size_t)t * (NHEADS * HDIM));
        const v4f    vv = __builtin_nontemporal_load((const v4f*)(vr + lane * 4));
        const float  p  = s_s[wave * 64 + i];
        acc += p * vv;
    }
    acc_s[wave * HDIM + lane * 4 + 0] = acc[0];
    acc_s[wave * HDIM + lane * 4 + 1] = acc[1];
    acc_s[wave * HDIM + lane * 4 + 2] = acc[2];
    acc_s[wave * HDIM + lane * 4 + 3] = acc[3];
    __syncthreads();

    const size_t idx = ((size_t)(b * NHEADS + h) * NCHUNK + c) * 130;
    if (tid < HDIM) {
        float tot = 0.0f;
        for (int w = 0; w < 8; ++w) tot += acc_s[w * HDIM + tid];
        part[idx + tid] = tot;
    }
    if (tid == 128) part[idx + 128] = mshare;
    if (tid == 129) part[idx + 129] = lshare;
}

// ---------------------------------------------------------------------------
// Kernel 4: combine chunk partials -> ws_attn [b][h*128+d].
// ---------------------------------------------------------------------------
__global__ void attn_reduce(const float* __restrict__ part,
                            float* __restrict__ attn) {
    const int bh = blockIdx.x;
    const int d  = threadIdx.x;
    float M = -INFINITY;
    for (int c = 0; c < NCHUNK; ++c)
        M = fmaxf(M, part[((size_t)bh * NCHUNK + c) * 130 + 128]);
    float L = 0.0f, val = 0.0f;
    for (int c = 0; c < NCHUNK; ++c) {
        const size_t o  = ((size_t)bh * NCHUNK + c) * 130;
        const float  sc = __expf(part[o + 128] - M);
        L   += part[o + 129] * sc;
        val += part[o + d] * sc;
    }
    attn[(size_t)bh * HDIM + d] = val / L;
}

// ---------------------------------------------------------------------------
// Kernel 5: output projection partials (attn @ wo^T), K split in half.
// Grid: 2 khalf x 32 nblk = 64 blocks, 256 threads.
// ---------------------------------------------------------------------------
__global__ void wo_gemm(const float* __restrict__ a,
                        const float* __restrict__ wo,
                        float* __restrict__ wop) {
    const int wg    = blockIdx.x;
    const int khalf = wg >> 5;
    const int nblk  = wg & 31;
    const int wave  = threadIdx.x >> 5;
    const int lane  = threadIdx.x & 31;
    const int n0    = nblk * 128 + wave * 16;
    const int half  = lane >> 4;
    const int mr    = lane & 15;
    const int k0    = khalf * KHALF;

    v8f acc = gemm_strip(a + (size_t)mr * EMB + k0 + 4 * half,
                         wo + (size_t)(n0 + mr) * EMB + k0 + 4 * half);

    float* dst = wop + (size_t)khalf * (BSZ * EMB);
    const int n = n0 + mr;
#pragma unroll
    for (int j = 0; j < 8; ++j) {
        const int M = j + 8 * half;
        dst[(size_t)M * EMB + n] = acc[j];
    }
}

// ---------------------------------------------------------------------------
// Kernel 6: sum the two K-half partials -> d_out.
// ---------------------------------------------------------------------------
__global__ void wo_reduce(const float* __restrict__ wop,
                          float* __restrict__ out) {
    const int e = blockIdx.x * 256 + threadIdx.x;   // 0 .. 65535
    out[e] = wop[e] + wop[BSZ * EMB + e];
}

// ---------------------------------------------------------------------------
// Launcher. Workspace layout (floats):
//   [0      ,  65536) ws_q          [65536 , 131072) ws_k
//   [131072 , 196608) ws_v          [196608, 262144) ws_attn
//   [262144 , 794624) ws_part  (4096 x 130)
//   [794624 ,1187840) ws_qkvp (3 x 2 x 16 x 4096)
//   [1187840,1318912) ws_wop  (2 x 16 x 4096)     total ~5.0 MB
// ---------------------------------------------------------------------------
extern "C" void kernel_launch(void* const* d_in, const int* in_sizes, int n_in,
                              void* d_out, int out_size, void* d_ws, size_t ws_size,
                              hipStream_t stream) {
    const float* x    = (const float*)d_in[0];
    const float* fcos = (const float*)d_in[1];
    const float* fsin = (const float*)d_in[2];
    const float* ck   = (const float*)d_in[3];
    const float* cv   = (const float*)d_in[4];
    const float* wq   = (const float*)d_in[5];
    const float* wk   = (const float*)d_in[6];
    const float* wv   = (const float*)d_in[7];
    const float* wo   = (const float*)d_in[8];
    const int*   sp   = (const int*)d_in[9];

    float* ws      = (float*)d_ws;
    float* ws_q    = ws;
    float* ws_k    = ws + 65536;
    float* ws_v    = ws + 131072;
    float* ws_attn = ws + 196608;
    float* ws_part = ws + 262144;
    float* ws_qkvp = ws + 794624;
    float* ws_wop  = ws + 1187840;
    float* out     = (float*)d_out;

    qkv_gemm<<<192, 256, 0, stream>>>(x, wq, wk, wv, ws_qkvp);
    qkv_rope_reduce<<<384, 256, 0, stream>>>(ws_qkvp, fcos, fsin, ws_q, ws_k, ws_v);
    attn_partial<<<BSZ * NHEADS * NCHUNK, 256, 0, stream>>>(ck, cv, ws_q, ws_k,
                                                            ws_v, ws_part, sp);
    attn_reduce<<<BSZ * NHEADS, HDIM, 0, stream>>>(ws_part, ws_attn);
    wo_gemm<<<64, 256, 0, stream>>>(ws_attn, wo, ws_wop);
    wo_reduce<<<256, 256, 0, stream>>>(ws_wop, out);
}